// ModelNew_3556232922036
// MI455X (gfx1250) — compile-verified
//
#include <hip/hip_runtime.h>
#include <math.h>

// ---------------------------------------------------------------------------
// Fused ConvTranspose2d(32->64,k5,s1,p1) + affine-BN + tanh + maxpool2x2 +
// per-pixel GroupNorm(8 groups) for MI455X (gfx1250).
//
// Implicit GEMM on V_WMMA_F32_16X16X4_F32 (f32-exact, compute-bound problem:
// ~55 GFLOP vs ~100MB traffic -> ~550 FLOP/byte).
//   M = Cout = 64 (4 WMMA M-tiles)
//   N = 256  = one 16x16 conv-output patch (wave w owns rows 2w, 2w+1)
//   K = 800  = Cin*25, tap-major (K = tap*32 + ci)
//
// LDS (dynamic, ~251.5 KB of the 320 KB WGP LDS), K-PAIR INTERLEAVED so every
// WMMA fragment is one ds_load_b64 into an even-aligned VGPR pair (no
// v_mov shuffling in the hot loop):
//   Wp[(tap*16+cipair)*64+co] : float2 {A[co][k], A[co][k+1]}      (200 KB)
//   Xp[cipair][20*20]         : float2 {x[2p],   x[2p+1]}          ( 51 KB)
//   per-channel fused affine constants                             (  1 KB)
//
// Epilogue: D-tile layout gives each lane whole 8-channel GN groups ->
// GroupNorm is lane-local; 2x2 maxpool = in-lane fmax + one shfl_xor.
// ---------------------------------------------------------------------------

typedef __attribute__((ext_vector_type(2))) float v2f;
typedef __attribute__((ext_vector_type(8))) float v8f;

#define CIN      32
#define COUT     64
#define HH       128
#define WW       128
#define POOLED   65
#define XPP      404                 // pairs per plane-pair (400 used + 4 pad)
#define WL_SIZE  (25*16*64*2)        // 51200 floats (25600 float2 pairs)
#define XS_SIZE  (16*XPP*2)          // 12928 floats (6464 float2 pairs)
#define SMEM_FLOATS (WL_SIZE + XS_SIZE + 4*64)

extern "C" __global__ __launch_bounds__(256)
void fused_convt_gn_kernel(const float* __restrict__ X,
                           const float* __restrict__ Wg,
                           const float* __restrict__ Bias,
                           const float* __restrict__ BnW,
                           const float* __restrict__ BnB,
                           const float* __restrict__ GnW,
                           const float* __restrict__ GnB,
                           float* __restrict__ Out)
{
    extern __shared__ float smem[];
    float* wl  = smem;                       // pair-interleaved weights
    float* xs  = smem + WL_SIZE;             // pair-interleaved halo patch
    float* sS  = xs + XS_SIZE;               // bn_w / sqrt(1+eps)
    float* tS  = sS + 64;                    // bias*s + bn_b
    float* gwS = tS + 64;
    float* gbS = gwS + 64;

    const int tid = threadIdx.x;
    const int bb  = blockIdx.z;
    const int oh0 = 16 * blockIdx.y;         // conv-output patch origin
    const int ow0 = 16 * blockIdx.x;

    // ---- stage flipped/transposed weights, K-pair interleaved -------------
    // wl[ ((tap*16 + p)*64 + co)*2 + e ] = A[co][tap*32 + 2p + e]
    for (int idx = tid; idx < WL_SIZE; idx += 256) {
        const int e   = idx & 1;
        const int co  = (idx >> 1) & 63;
        const int p   = (idx >> 7) & 15;
        const int tap = idx >> 11;
        const int ci  = 2 * p + e;
        const int kh  = tap / 5, kw = tap - 5 * kh;
        wl[idx] = Wg[(ci * COUT + co) * 25 + (4 - kh) * 5 + (4 - kw)];
    }
    // ---- stage 20x20 halo patch, plane-pair interleaved, zero-padded ------
    const float* Xb = X + (size_t)bb * CIN * HH * WW;
    for (int idx = tid; idx < CIN * 400; idx += 256) {
        const int ci  = idx / 400;
        const int rem = idx - ci * 400;
        const int r   = rem / 20, c = rem - 20 * r;
        const int gy  = oh0 - 3 + r, gx = ow0 - 3 + c;
        float v = 0.0f;
        if ((unsigned)gy < 128u && (unsigned)gx < 128u)
            v = Xb[ci * (HH * WW) + gy * WW + gx];
        xs[((ci >> 1) * XPP + rem) * 2 + (ci & 1)] = v;
    }
    if (tid < 64) {
        const float s = BnW[tid] * 0.9999950000374996f;   // 1/sqrt(1+1e-5)
        sS[tid]  = s;
        tS[tid]  = Bias[tid] * s + BnB[tid];
        gwS[tid] = GnW[tid];
        gbS[tid] = GnB[tid];
    }
    __syncthreads();

    const int wave = tid >> 5;
    const int lane = tid & 31;
    const int half = lane >> 4;     // selects K-pair within the K-quad
    const int ln   = lane & 15;     // M index (A) / N index (B,C)

    v8f acc[4][2] = {};             // [M-tile][row-within-wave]

    const v2f* Wp = (const v2f*)wl; // [ (tap*16 + p)*64 + co ]
    const v2f* Xq = (const v2f*)xs; // [ p*XPP + r*20 + c ]
    const int  xroff = (2 * wave) * 20 + ln;   // this wave's base pixel

#pragma unroll 1
    for (int tap = 0; tap < 25; ++tap) {
        const int kh = tap / 5, kw = tap - 5 * kh;
        const v2f* wbase = Wp + (tap * 16 + half) * 64 + ln;
        const v2f* xbase = Xq + half * XPP + (kh * 20 + kw) + xroff;
        for (int ci0 = 0; ci0 < 32; ci0 += 4) {
            const int p2 = ci0 >> 1;            // K-quad -> pair index
            // A fragments: one b64 each, ISA K-split handled by `half`
            const v2f a0 = wbase[p2 * 64 + 0];
            const v2f a1 = wbase[p2 * 64 + 16];
            const v2f a2 = wbase[p2 * 64 + 32];
            const v2f a3 = wbase[p2 * 64 + 48];
            // B fragments: one b64 each, rows 2w and 2w+1
            const v2f b0 = xbase[p2 * XPP];
            const v2f b1 = xbase[p2 * XPP + 20];
            acc[0][0] = __builtin_amdgcn_wmma_f32_16x16x4_f32(false, a0, false, b0, (short)0, acc[0][0], false, false);
            acc[1][0] = __builtin_amdgcn_wmma_f32_16x16x4_f32(false, a1, false, b0, (short)0, acc[1][0], false, false);
            acc[2][0] = __builtin_amdgcn_wmma_f32_16x16x4_f32(false, a2, false, b0, (short)0, acc[2][0], false, false);
            acc[3][0] = __builtin_amdgcn_wmma_f32_16x16x4_f32(false, a3, false, b0, (short)0, acc[3][0], false, false);
            acc[0][1] = __builtin_amdgcn_wmma_f32_16x16x4_f32(false, a0, false, b1, (short)0, acc[0][1], false, false);
            acc[1][1] = __builtin_amdgcn_wmma_f32_16x16x4_f32(false, a1, false, b1, (short)0, acc[1][1], false, false);
            acc[2][1] = __builtin_amdgcn_wmma_f32_16x16x4_f32(false, a2, false, b1, (short)0, acc[2][1], false, false);
            acc[3][1] = __builtin_amdgcn_wmma_f32_16x16x4_f32(false, a3, false, b1, (short)0, acc[3][1], false, false);
        }
    }

    // ---- fused epilogue: affine-BN + tanh + 2x2 maxpool + per-pixel GN ----
    const int  prow = 8 * blockIdx.y + wave;
    const int  pcol = 8 * blockIdx.x + (ln >> 1);
    const bool doStore = ((ln & 1) == 0) && (prow < POOLED) && (pcol < POOLED);
    float* outB = Out + (size_t)bb * COUT * POOLED * POOLED;

#pragma unroll
    for (int Mt = 0; Mt < 4; ++Mt) {
        float pooled[8];
        float sum = 0.0f, ssq = 0.0f;
#pragma unroll
        for (int r = 0; r < 8; ++r) {
            const int   co = Mt * 16 + 8 * half + r;
            const float sc = sS[co], tc = tS[co];
            const float y0 = tanhf(acc[Mt][0][r] * sc + tc);
            const float y1 = tanhf(acc[Mt][1][r] * sc + tc);
            float v = fmaxf(y0, y1);                 // pool rows (in-lane)
            v = fmaxf(v, __shfl_xor(v, 1, 32));      // pool cols (lane pair)
            pooled[r] = v;
            sum += v; ssq += v * v;
        }
        const float mu  = sum * 0.125f;
        const float var = ssq * 0.125f - mu * mu;
        const float inv = rsqrtf(var + 1e-5f);
        if (doStore) {
#pragma unroll
            for (int r = 0; r < 8; ++r) {
                const int co = Mt * 16 + 8 * half + r;
                outB[(co * POOLED + prow) * POOLED + pcol] =
                    (pooled[r] - mu) * inv * gwS[co] + gbS[co];
            }
        }
    }
}

extern "C" void kernel_launch(void* const* d_in, const int* in_sizes, int n_in,
                              void* d_out, int out_size, void* d_ws, size_t ws_size,
                              hipStream_t stream) {
    (void)in_sizes; (void)n_in; (void)out_size; (void)d_ws; (void)ws_size;
    const float* x   = (const float*)d_in[0];
    const float* w   = (const float*)d_in[1];
    const float* b   = (const float*)d_in[2];
    const float* bnw = (const float*)d_in[3];
    const float* bnb = (const float*)d_in[4];
    const float* gnw = (const float*)d_in[5];
    const float* gnb = (const float*)d_in[6];
    float* out = (float*)d_out;

    dim3 grid((POOLED + 7) / 8, (POOLED + 7) / 8, 32);   // 9 x 9 x 32
    dim3 block(256);
    const size_t shmem = (size_t)SMEM_FLOATS * sizeof(float);  // ~251.5 KB
    fused_convt_gn_kernel<<<grid, block, shmem, stream>>>(x, w, b, bnw, bnb,
                                                          gnw, gnb, out);
}